// Model_48103633715264
// MI455X (gfx1250) — compile-verified
//
#include <hip/hip_runtime.h>

#define H1 8
#define C1 32
#define C2 32
#define NEG_SLOPE 0.2f

typedef __attribute__((ext_vector_type(2))) float v2f;
typedef __attribute__((ext_vector_type(8))) float v8f;

// ---- helpers -------------------------------------------------------------

// monotone float -> uint key for atomicMax-based segment max
__device__ __forceinline__ unsigned fkey(float f) {
    unsigned u = __float_as_uint(f);
    return (u & 0x80000000u) ? ~u : (u | 0x80000000u);
}
__device__ __forceinline__ float funkey(unsigned k) {
    return (k & 0x80000000u) ? __uint_as_float(k ^ 0x80000000u)
                             : __uint_as_float(~k);
}
__device__ __forceinline__ float lrelu(float e) {
    return e > 0.f ? e : NEG_SLOPE * e;
}
__device__ __forceinline__ void edge_sd(const int* __restrict__ ei, int i, int E,
                                        int& s, int& d) {
    if (i < E) { s = ei[i]; d = ei[E + i]; }
    else       { s = i - E; d = i - E; }          // appended self-loops
}

// ---- kernels -------------------------------------------------------------

__global__ void k_zero(float* __restrict__ p, size_t n) {
    size_t i = (size_t)blockIdx.x * blockDim.x + threadIdx.x;
    size_t stride = (size_t)gridDim.x * blockDim.x;
    for (; i < n; i += stride) p[i] = 0.f;
}

// S1[h] = sum_c W1[h,c]*att_src1[h,c]; D1 likewise (layer-1 rank-1 collapse)
__global__ void k_prep(const float* __restrict__ W1, const float* __restrict__ as1,
                       const float* __restrict__ ad1,
                       float* __restrict__ S1, float* __restrict__ D1) {
    int h = threadIdx.x;
    if (h < H1) {
        float s = 0.f, d = 0.f;
        for (int c = 0; c < C1; ++c) {
            float w = W1[h * C1 + c];
            s += w * as1[h * C1 + c];
            d += w * ad1[h * C1 + c];
        }
        S1[h] = s;
        D1[h] = d;
    }
}

__global__ void k_e1_max(const int* __restrict__ ei, const float* __restrict__ x,
                         const float* __restrict__ S1, const float* __restrict__ D1,
                         unsigned* __restrict__ m1, int E, int ET) {
    int i = blockIdx.x * blockDim.x + threadIdx.x;
    if (i >= ET) return;
    int s, d; edge_sd(ei, i, E, s, d);
    float xs = x[s], xd = x[d];
#pragma unroll
    for (int h = 0; h < H1; ++h) {
        float e = lrelu(xs * S1[h] + xd * D1[h]);
        atomicMax(&m1[d * H1 + h], fkey(e));
    }
}

__global__ void k_e1_den(const int* __restrict__ ei, const float* __restrict__ x,
                         const float* __restrict__ S1, const float* __restrict__ D1,
                         const unsigned* __restrict__ m1, float* __restrict__ den1,
                         int E, int ET) {
    int i = blockIdx.x * blockDim.x + threadIdx.x;
    if (i >= ET) return;
    int s, d; edge_sd(ei, i, E, s, d);
    float xs = x[s], xd = x[d];
#pragma unroll
    for (int h = 0; h < H1; ++h) {
        float e = lrelu(xs * S1[h] + xd * D1[h]);
        float ex = expf(e - funkey(m1[d * H1 + h]));
        atomicAdd(&den1[d * H1 + h], ex);
    }
}

// t[d,h] += x[src] * alpha  (rank-1 message aggregation)
__global__ void k_e1_acc(const int* __restrict__ ei, const float* __restrict__ x,
                         const float* __restrict__ S1, const float* __restrict__ D1,
                         const unsigned* __restrict__ m1, const float* __restrict__ den1,
                         float* __restrict__ tacc, int E, int ET) {
    int i = blockIdx.x * blockDim.x + threadIdx.x;
    if (i >= ET) return;
    int s, d; edge_sd(ei, i, E, s, d);
    float xs = x[s], xd = x[d];
#pragma unroll
    for (int h = 0; h < H1; ++h) {
        float e = lrelu(xs * S1[h] + xd * D1[h]);
        float ex = expf(e - funkey(m1[d * H1 + h]));
        float alpha = ex / (den1[d * H1 + h] + 1e-16f);
        atomicAdd(&tacc[d * H1 + h], xs * alpha);
    }
}

// h2 = relu(W1[k]*t[n,k/32] + b1[k])  @  W2[256,32], via V_WMMA_F32_16X16X4_F32.
// One wave -> one 16-row tile, both 16-col tiles. A-tile generated on the fly.
__global__ void k_gemm(const float* __restrict__ tacc, const float* __restrict__ W1,
                       const float* __restrict__ b1, const float* __restrict__ W2,
                       float* __restrict__ h2, int N, int ntiles) {
    __shared__ float sW2[256 * 32];
    __shared__ float sW1[256];
    __shared__ float sb1[256];
    for (int i = threadIdx.x; i < 256 * 32; i += blockDim.x) sW2[i] = W2[i];
    for (int i = threadIdx.x; i < 256; i += blockDim.x) { sW1[i] = W1[i]; sb1[i] = b1[i]; }
    __syncthreads();

    int warp = threadIdx.x >> 5;
    int lane = threadIdx.x & 31;
    int tile = blockIdx.x * (blockDim.x >> 5) + warp;
    if (tile >= ntiles) return;

    int row0  = tile * 16;
    int mloc  = lane & 15;           // M (for A) or N (for B/D) within tile
    int khalf = (lane >> 4) << 1;    // lanes 16-31 hold K=2,3
    int row   = row0 + mloc;

    v8f acc0 = {};
    v8f acc1 = {};

#pragma unroll 1
    for (int h = 0; h < 8; ++h) {
        float th = (row < N) ? tacc[row * 8 + h] : 0.f;
        int kh = h * 32;
#pragma unroll
        for (int kk = 0; kk < 32; kk += 4) {
            int kb = kh + kk + khalf;
            // A (16x4 f32): lanes 0-15 hold K=kb,kb+1 ; lanes 16-31 K=kb(+2),kb+1
            float a0 = sW1[kb] * th + sb1[kb];
            float a1 = sW1[kb + 1] * th + sb1[kb + 1];
            v2f a;
            a.x = a0 > 0.f ? a0 : 0.f;     // relu -> layer-2 input feature
            a.y = a1 > 0.f ? a1 : 0.f;
            // B (4x16 f32): row K striped across lanes
            v2f b0, b1v;
            b0.x  = sW2[kb * 32 + mloc];
            b0.y  = sW2[(kb + 1) * 32 + mloc];
            b1v.x = sW2[kb * 32 + 16 + mloc];
            b1v.y = sW2[(kb + 1) * 32 + 16 + mloc];
            acc0 = __builtin_amdgcn_wmma_f32_16x16x4_f32(
                       false, a, false, b0, (short)0, acc0, false, false);
            acc1 = __builtin_amdgcn_wmma_f32_16x16x4_f32(
                       false, a, false, b1v, (short)0, acc1, false, false);
        }
    }

    // D layout: VGPR g -> row g (lanes 0-15) / row g+8 (lanes 16-31), col = lane&15
    int rbase = row0 + ((lane >> 4) << 3);
#pragma unroll
    for (int g = 0; g < 8; ++g) {
        int r = rbase + g;
        if (r < N) {
            h2[r * 32 + mloc]      = acc0[g];
            h2[r * 32 + 16 + mloc] = acc1[g];
        }
    }
}

__global__ void k_att2(const float* __restrict__ h2, const float* __restrict__ as2,
                       const float* __restrict__ ad2, float* __restrict__ a_s,
                       float* __restrict__ a_d, int N) {
    int n = blockIdx.x * blockDim.x + threadIdx.x;
    if (n >= N) return;
    float s = 0.f, d = 0.f;
#pragma unroll
    for (int c = 0; c < C2; ++c) {
        float v = h2[n * C2 + c];
        s += v * as2[c];
        d += v * ad2[c];
    }
    a_s[n] = s;
    a_d[n] = d;
}

__global__ void k_e2_max(const int* __restrict__ ei, const float* __restrict__ a_s,
                         const float* __restrict__ a_d, unsigned* __restrict__ m2,
                         int E, int ET) {
    int i = blockIdx.x * blockDim.x + threadIdx.x;
    if (i >= ET) return;
    int s, d; edge_sd(ei, i, E, s, d);
    float e = lrelu(a_s[s] + a_d[d]);
    atomicMax(&m2[d], fkey(e));
}

__global__ void k_e2_den(const int* __restrict__ ei, const float* __restrict__ a_s,
                         const float* __restrict__ a_d, const unsigned* __restrict__ m2,
                         float* __restrict__ den2, int E, int ET) {
    int i = blockIdx.x * blockDim.x + threadIdx.x;
    if (i >= ET) return;
    int s, d; edge_sd(ei, i, E, s, d);
    float e = lrelu(a_s[s] + a_d[d]);
    atomicAdd(&den2[d], expf(e - funkey(m2[d])));
}

__global__ void k_e2_acc(const int* __restrict__ ei, const float* __restrict__ a_s,
                         const float* __restrict__ a_d, const unsigned* __restrict__ m2,
                         const float* __restrict__ den2, const float* __restrict__ h2,
                         float* __restrict__ out2, int E, int ET) {
    int i = blockIdx.x * blockDim.x + threadIdx.x;
    if (i >= ET) return;
    int s, d; edge_sd(ei, i, E, s, d);
    float e = lrelu(a_s[s] + a_d[d]);
    float alpha = expf(e - funkey(m2[d])) / (den2[d] + 1e-16f);
#pragma unroll
    for (int c = 0; c < C2; ++c)
        atomicAdd(&out2[d * C2 + c], h2[s * C2 + c] * alpha);
}

__global__ void k_pool(const float* __restrict__ out2, const float* __restrict__ b2,
                       const int* __restrict__ batch, float* __restrict__ sums,
                       float* __restrict__ cnt, int N) {
    int n = blockIdx.x * blockDim.x + threadIdx.x;
    if (n >= N) return;
    int g = batch[n];
#pragma unroll
    for (int c = 0; c < C2; ++c) {
        float v = out2[n * C2 + c] + b2[c];
        v = v > 0.f ? v : 0.f;                 // relu after gat2
        atomicAdd(&sums[g * C2 + c], v);
    }
    atomicAdd(&cnt[g], 1.f);
}

__global__ void k_final(const float* __restrict__ sums, const float* __restrict__ cnt,
                        const float* __restrict__ lw, const float* __restrict__ lb,
                        float* __restrict__ out, int G) {
    int g = blockIdx.x * blockDim.x + threadIdx.x;
    if (g >= G) return;
    float inv = 1.f / fmaxf(cnt[g], 1.f);
    float acc = 0.f;
#pragma unroll
    for (int c = 0; c < C2; ++c)
        acc += sums[g * C2 + c] * inv * lw[c];
    out[g] = acc + lb[0];
}

// ---- launcher ------------------------------------------------------------

extern "C" void kernel_launch(void* const* d_in, const int* in_sizes, int n_in,
                              void* d_out, int out_size, void* d_ws, size_t ws_size,
                              hipStream_t stream) {
    const float* x    = (const float*)d_in[0];
    const int*   ei   = (const int*)d_in[1];
    const int*   batch= (const int*)d_in[2];
    const float* W1   = (const float*)d_in[3];
    const float* as1  = (const float*)d_in[4];
    const float* ad1  = (const float*)d_in[5];
    const float* b1   = (const float*)d_in[6];
    const float* W2   = (const float*)d_in[7];
    const float* as2  = (const float*)d_in[8];
    const float* ad2  = (const float*)d_in[9];
    const float* b2   = (const float*)d_in[10];
    const float* lw   = (const float*)d_in[11];
    const float* lb   = (const float*)d_in[12];
    float* out = (float*)d_out;

    const int N  = in_sizes[0];        // x is [N,1]
    const int E  = in_sizes[1] / 2;    // edge_index is [2,E]
    const int G  = out_size;           // output [G,1]
    const int ET = E + N;              // edges + self loops

    // workspace layout (4-byte units)
    float* base = (float*)d_ws;
    size_t off = 0;
    float*    S1   = base + off; off += 8;
    float*    D1   = base + off; off += 8;
    unsigned* m1   = (unsigned*)(base + off); off += (size_t)N * H1;
    float*    den1 = base + off; off += (size_t)N * H1;
    float*    tacc = base + off; off += (size_t)N * H1;
    float*    h2   = base + off; off += (size_t)N * C2;
    float*    a_s  = base + off; off += (size_t)N;
    float*    a_d  = base + off; off += (size_t)N;
    unsigned* m2   = (unsigned*)(base + off); off += (size_t)N;
    float*    den2 = base + off; off += (size_t)N;
    float*    out2 = base + off; off += (size_t)N * C2;
    float*    sums = base + off; off += (size_t)G * C2;
    float*    cnt  = base + off; off += (size_t)G;
    size_t total = off;

    k_zero<<<2048, 256, 0, stream>>>(base, total);
    k_prep<<<1, 32, 0, stream>>>(W1, as1, ad1, S1, D1);

    int eb = (ET + 255) / 256;
    k_e1_max<<<eb, 256, 0, stream>>>(ei, x, S1, D1, m1, E, ET);
    k_e1_den<<<eb, 256, 0, stream>>>(ei, x, S1, D1, m1, den1, E, ET);
    k_e1_acc<<<eb, 256, 0, stream>>>(ei, x, S1, D1, m1, den1, tacc, E, ET);

    int ntiles = (N + 15) / 16;
    int wgs = (ntiles + 7) / 8;        // 8 waves (256 thr) per workgroup
    k_gemm<<<wgs, 256, 0, stream>>>(tacc, W1, b1, W2, h2, N, ntiles);

    int nb = (N + 255) / 256;
    k_att2<<<nb, 256, 0, stream>>>(h2, as2, ad2, a_s, a_d, N);
    k_e2_max<<<eb, 256, 0, stream>>>(ei, a_s, a_d, m2, E, ET);
    k_e2_den<<<eb, 256, 0, stream>>>(ei, a_s, a_d, m2, den2, E, ET);
    k_e2_acc<<<eb, 256, 0, stream>>>(ei, a_s, a_d, m2, den2, h2, out2, E, ET);

    k_pool<<<nb, 256, 0, stream>>>(out2, b2, batch, sums, cnt, N);
    k_final<<<1, 256, 0, stream>>>(sums, cnt, lw, lb, out, G);
}